// RSA_layer_51488067945057
// MI455X (gfx1250) — compile-verified
//
#include <hip/hip_runtime.h>
#include <cstdint>

typedef __attribute__((ext_vector_type(2))) float v2f;
typedef __attribute__((ext_vector_type(8))) float v8f;

// Problem constants (B, U, W) = (64, 128, 128); U == W so flip == new_state.
#define NB    64
#define DIM   128
#define PAD   130   // LDS row stride in floats -> conflict-free column walks (bank = (2j+u)%64)

__global__ __launch_bounds__(256, 1)
void rsa_layer_kernel(const float* __restrict__ input,   // (64,128)
                      const float* __restrict__ state,   // (64,128,128)
                      const float* __restrict__ w,       // (257,128) row-major
                      const float* __restrict__ bias,    // (128,)
                      float* __restrict__ out)           // (64,128)
{
    __shared__ float F[DIM * PAD];      // flip[b]  (row j, col u)
    __shared__ float S[DIM * PAD];      // H, then sim (row j, col v)
    __shared__ float dotArr[DIM];

    const int b   = blockIdx.x;
    const int tid = threadIdx.x;

    // Warm L2/WGP$ for the shared weight matrix (all 64 blocks reuse it)
    __builtin_prefetch(w, 0, 1);

    // ---- Phase 0: flip[b] = [state[b,:,1:] | input[b,:,None]] -> LDS, via async-to-LDS ----
    // The +1 column shift leaves only 4B alignment, so use B32 async copies (ASYNCcnt path,
    // no VGPR round-trip). LDS byte address = low 32 bits of the generic pointer.
    const float* stateB = state + (size_t)b * DIM * DIM;
    for (int idx = tid; idx < DIM * DIM; idx += 256) {
        const int j = idx >> 7;
        const int u = idx & 127;
        const float*   gsrc   = (u < DIM - 1) ? (stateB + j * DIM + u + 1)
                                              : (input + b * DIM + j);
        const uint32_t ldsOff = (uint32_t)(uintptr_t)(&F[j * PAD + u]);
        asm volatile("global_load_async_to_lds_b32 %0, %1, off"
                     :: "v"(ldsOff), "v"(gsrc) : "memory");
    }
    asm volatile("s_wait_asynccnt 0x0" ::: "memory");
    __syncthreads();

    // ---- Phase 1: H = F @ w_hi  (M=N=K=128) via V_WMMA_F32_16X16X4_F32 ----
    // k outer / nb inner: 8 persistent accumulators, A fragment reused 8x per k step,
    // low register pressure (no spills).
    const int wave = tid >> 5;          // 8 waves -> 8 M-tile rows
    const int lane = tid & 31;
    const int half = lane >> 4;         // A layout: lanes 0-15 hold K={0,1}, 16-31 hold K={2,3}
    const int l15  = lane & 15;
    const int rowA = wave * 16 + l15;

    v8f acc[8] = {};
    #pragma unroll 2
    for (int k = 0; k < 32; ++k) {
        const int kk = k * 4 + half * 2;
        v2f a;
        a.x = F[rowA * PAD + kk];       // ds_load_b64 (consecutive pair)
        a.y = F[rowA * PAD + kk + 1];
        #pragma unroll
        for (int nb = 0; nb < 8; ++nb) {
            const int col = nb * 16 + l15;
            v2f bf;
            bf.x = w[kk * DIM + col];            // w_hi = w[0:128], L2 resident
            bf.y = w[(kk + 1) * DIM + col];
            acc[nb] = __builtin_amdgcn_wmma_f32_16x16x4_f32(
                false, a, false, bf, (short)0, acc[nb], false, false);
        }
    }
    // C/D layout: VGPR r, lanes 0-15 -> M = 16*wave + r, lanes 16-31 -> M = 16*wave + r + 8
    #pragma unroll
    for (int nb = 0; nb < 8; ++nb) {
        const int col = nb * 16 + l15;
        #pragma unroll
        for (int r = 0; r < 8; ++r)
            S[(wave * 16 + r + 8 * half) * PAD + col] = acc[nb][r];
    }
    __syncthreads();

    // ---- Phase 2: dot[j] = F[127,:]·F[j,:]  and  g[v] = F[127,:] @ w_hj[:,v] ----
    float g = 0.f;
    if (tid < DIM) {
        float d = 0.f;
        for (int u = 0; u < DIM; ++u) {
            const float x = F[(DIM - 1) * PAD + u];     // LDS broadcast
            d += x * F[tid * PAD + u];
            g += x * w[(DIM + u) * DIM + tid];          // w_hj = w[128:256]
        }
        dotArr[tid] = d;
    }
    __syncthreads();

    // ---- Phase 3: column-wise softmax over j, fused with weighted reduction ----
    if (tid < DIM) {
        const int   v  = tid;
        const float wd = w[2 * DIM * DIM + v];          // w_dot = w[256]
        const float gb = g + bias[v];

        float m = -3.4e38f;
        for (int j = 0; j < DIM; ++j) {
            const float s = S[j * PAD + v] + gb + dotArr[j] * wd;
            S[j * PAD + v] = s;
            m = fmaxf(m, s);
        }
        float sum = 0.f, cacc = 0.f;
        for (int j = 0; j < DIM; ++j) {
            const float e = __expf(S[j * PAD + v] - m);
            sum  += e;
            cacc += F[j * PAD + v] * e;                 // c[v] = sum_j F[j,v] * a[j,v]
        }
        out[b * DIM + v] = cacc / sum;
    }
}

extern "C" void kernel_launch(void* const* d_in, const int* in_sizes, int n_in,
                              void* d_out, int out_size, void* d_ws, size_t ws_size,
                              hipStream_t stream) {
    const float* input = (const float*)d_in[0];   // (64,128)
    const float* state = (const float*)d_in[1];   // (64,128,128)
    const float* w     = (const float*)d_in[2];   // (257,128)
    const float* bias  = (const float*)d_in[3];   // (128,)
    float*       out   = (float*)d_out;           // (64,128)
    rsa_layer_kernel<<<NB, 256, 0, stream>>>(input, state, w, bias, out);
}